// ViT_Detector_592705487431
// MI455X (gfx1250) — compile-verified
//
#include <hip/hip_runtime.h>
#include <math.h>

// ---------------------------------------------------------------------------
// ViT detector forward on gfx1250 (CDNA5), fp32 WMMA (V_WMMA_F32_16X16X4_F32)
// + async global->LDS staging (GLOBAL_LOAD_ASYNC_TO_LDS_B128 / ASYNCcnt).
// ---------------------------------------------------------------------------

typedef float v2f __attribute__((ext_vector_type(2)));
typedef float v8f __attribute__((ext_vector_type(8)));
typedef int   v4i __attribute__((ext_vector_type(4)));

#define WMMA_F32(a, b, c) \
  __builtin_amdgcn_wmma_f32_16x16x4_f32(false, (a), false, (b), (short)0, (c), false, false)

// Async global->LDS copy (CDNA5 GLOBAL_LOAD_ASYNC_TO_LDS_B128), guarded so a
// toolchain without the builtin falls back to the sync VGPR round-trip.
// Builtin params are (v4i addrspace(1)* src, v4i addrspace(3)* dst, imm off, imm cpol).
#if defined(__HIP_DEVICE_COMPILE__) && __has_builtin(__builtin_amdgcn_global_load_async_to_lds_b128)
#define HAS_ASYNC_LDS 1
#define ASYNC_CP16(gptr, lptr)                                              \
  __builtin_amdgcn_global_load_async_to_lds_b128(                           \
      (__attribute__((address_space(1))) v4i*)(gptr),                       \
      (__attribute__((address_space(3))) v4i*)(lptr), 0, 0)
#if __has_builtin(__builtin_amdgcn_s_wait_asynccnt)
#define ASYNC_WAIT() __builtin_amdgcn_s_wait_asynccnt(0)
#else
#define ASYNC_WAIT() asm volatile("s_wait_asynccnt 0" ::: "memory")
#endif
#else
#define HAS_ASYNC_LDS 0
#define ASYNC_CP16(gptr, lptr) \
  do { *(float4*)(lptr) = *(const float4*)(gptr); } while (0)
#define ASYNC_WAIT() do {} while (0)
#endif

#define N_LAYERS 4
#define N_HEADS  8
#define HIDDEN   256
#define FFWD     1024
#define SEQ      1024     // ROW*COL
#define BATCH    8
#define IN_CH    192
#define HEAD_D   32
#define NCOUT    85       // N_CLASSES + 5
#define MTOT     (BATCH * SEQ)   // 8192 rows

// ======================= patchify =========================================
__global__ __launch_bounds__(256) void patchify_kernel(
    float* __restrict__ out, const float* __restrict__ img) {
  int idx = blockIdx.x * blockDim.x + threadIdx.x;
  const int total = BATCH * SEQ * IN_CH;
  if (idx >= total) return;
  int ch = idx % 3;  int q = idx / 3;
  int kc = q % 8;    q /= 8;
  int kr = q % 8;    q /= 8;
  int c  = q % 32;   q /= 32;
  int r  = q % 32;   int b = q / 32;
  // x_image[b, r*8+kr, c*8+kc, ch], image is [B,256,256,3]
  size_t in = (((size_t)b * 256 + (size_t)r * 8 + kr) * 256 + (size_t)c * 8 + kc) * 3 + ch;
  out[idx] = img[in];
}

// ======================= pos add ==========================================
__global__ __launch_bounds__(256) void add_pos_kernel(
    float* __restrict__ out, const float* __restrict__ in,
    const float* __restrict__ pos) {
  int idx = blockIdx.x * blockDim.x + threadIdx.x;       // one float4 each
  const int per_batch = SEQ * HIDDEN / 4;                // 65536 float4
  const float4 a = ((const float4*)in)[idx];
  const float4 p = ((const float4*)pos)[idx % per_batch];
  float4 r;
  r.x = a.x + p.x; r.y = a.y + p.y; r.z = a.z + p.z; r.w = a.w + p.w;
  ((float4*)out)[idx] = r;
}

// ======================= residual + LayerNorm =============================
// out[row] = resid[row] + LN(x[row]) * gamma + beta   (row length 256)
__global__ __launch_bounds__(256) void ln_add_kernel(
    float* __restrict__ out, const float* __restrict__ resid,
    const float* __restrict__ x, const float* __restrict__ beta,
    const float* __restrict__ gamma) {
  const int wave = threadIdx.x / 32, lane = threadIdx.x % 32;
  const size_t row = (size_t)blockIdx.x * 8 + wave;
  const float* xr = x + row * HIDDEN;
  float v[8];
  float s = 0.f;
#pragma unroll
  for (int j = 0; j < 8; ++j) { v[j] = xr[lane + j * 32]; s += v[j]; }
#pragma unroll
  for (int off = 1; off < 32; off <<= 1) s += __shfl_xor(s, off, 32);
  const float mu = s * (1.f / 256.f);
  float var = 0.f;
#pragma unroll
  for (int j = 0; j < 8; ++j) { float d = v[j] - mu; var += d * d; }
#pragma unroll
  for (int off = 1; off < 32; off <<= 1) var += __shfl_xor(var, off, 32);
  const float rstd = rsqrtf(var * (1.f / 256.f) + 1e-6f);
#pragma unroll
  for (int j = 0; j < 8; ++j) {
    int c = lane + j * 32;
    out[row * HIDDEN + c] =
        resid[row * HIDDEN + c] + (v[j] - mu) * rstd * gamma[c] + beta[c];
  }
}

// ======================= generic WMMA GEMM ================================
// C[M,N] = A[M,K] @ B[K,N]; epilogues:
//   0 none | 1 *= scaleval | 2 relu(v+bias[col]) | 3 v+bias[col]
//   4 col<4 -> sigmoid(v) ; col>=4 -> v + bias[col-4]
// Block = 256 threads (8 waves); tile 64(M) x 128(N), BK=16.
// A/B tiles staged to LDS via GLOBAL_LOAD_ASYNC_TO_LDS_B128 when available.
__global__ __launch_bounds__(256) void gemm_wmma_kernel(
    const float* __restrict__ A, const float* __restrict__ B,
    float* __restrict__ C, int M, int N, int K,
    int epi, const float* __restrict__ bias, float scaleval) {
  __shared__ float sA[64][20];     // 80B row stride: 16-bank column reads, 16B aligned
  __shared__ float sB[16][136];    // 544B row stride: half-wave bank offset = 16

  const int tid  = threadIdx.x;
  const int wave = tid / 32, lane = tid % 32;
  const int l16  = lane % 16, half = lane / 16;
  const int bM = blockIdx.y * 64;
  const int bN = blockIdx.x * 128;
  const int wm = (wave / 4) * 32;          // 2x4 wave grid of 32x32 sub-tiles
  const int wn = (wave % 4) * 32;

  v8f acc[2][2] = {};
  const bool fastB = ((N & 3) == 0) && (bN + 128 <= N);

  const int arow = tid / 4, aseg = (tid % 4) * 4;
  const int brow = tid / 32, bc4a = (tid % 32) * 4;      // fastB: 2 segs/thread

  for (int k0 = 0; k0 < K; k0 += 16) {
    // ---- stage A tile 64x16 (async -> LDS) ----
    ASYNC_CP16(A + (size_t)(bM + arow) * K + k0 + aseg, &sA[arow][aseg]);
    // ---- stage B tile 16x128 ----
    if (fastB) {
      ASYNC_CP16(B + (size_t)(k0 + brow) * N + bN + bc4a, &sB[brow][bc4a]);
      ASYNC_CP16(B + (size_t)(k0 + 8 + brow) * N + bN + bc4a, &sB[8 + brow][bc4a]);
    } else {
#pragma unroll
      for (int it = 0; it < 8; ++it) {
        int s = tid + it * 256;                 // 0..2047
        int row = s / 128, col = s % 128;
        int gc = bN + col;
        sB[row][col] = (gc < N) ? B[(size_t)(k0 + row) * N + gc] : 0.f;
      }
    }
    if (k0 + 16 < K) {  // hint next tiles into cache (global_prefetch_b8)
      __builtin_prefetch(A + (size_t)(bM + arow) * K + k0 + 16 + aseg, 0, 1);
      __builtin_prefetch(B + (size_t)(k0 + 16 + (tid & 15)) * N + bN, 0, 1);
    }
    ASYNC_WAIT();            // own async copies complete (ASYNCcnt==0) ...
    __syncthreads();         // ... then all waves' tiles visible

#pragma unroll
    for (int kk = 0; kk < 4; ++kk) {
      const int kd = kk * 4 + half * 2;
      v2f a0, a1, b0, b1;
      a0.x = sA[wm + l16][kd];        a0.y = sA[wm + l16][kd + 1];
      a1.x = sA[wm + 16 + l16][kd];   a1.y = sA[wm + 16 + l16][kd + 1];
      b0.x = sB[kd][wn + l16];        b0.y = sB[kd + 1][wn + l16];
      b1.x = sB[kd][wn + 16 + l16];   b1.y = sB[kd + 1][wn + 16 + l16];
      acc[0][0] = WMMA_F32(a0, b0, acc[0][0]);
      acc[0][1] = WMMA_F32(a0, b1, acc[0][1]);
      acc[1][0] = WMMA_F32(a1, b0, acc[1][0]);
      acc[1][1] = WMMA_F32(a1, b1, acc[1][1]);
    }
    __syncthreads();
  }

  // ---- epilogue + store (C layout: vgpr i -> row i + 8*half, col l16) ----
#pragma unroll
  for (int mt = 0; mt < 2; ++mt)
#pragma unroll
    for (int nt = 0; nt < 2; ++nt)
#pragma unroll
      for (int i = 0; i < 8; ++i) {
        const int row = bM + wm + mt * 16 + i + 8 * half;
        const int col = bN + wn + nt * 16 + l16;
        if (col < N) {
          float v = acc[mt][nt][i];
          if (epi == 1)      v *= scaleval;
          else if (epi == 2) v = fmaxf(v + bias[col], 0.f);
          else if (epi == 3) v = v + bias[col];
          else if (epi == 4) v = (col < 4) ? 1.f / (1.f + __expf(-v))
                                           : v + bias[col - 4];
          C[(size_t)row * N + col] = v;
        }
      }
}

// ======================= flash attention ==================================
// grid.x = B*H (64), grid.y = S/128; each wave owns a 16-query tile.
// Q is pre-scaled by 1/sqrt(d). Q/K/V/O are [B,S,256] with head h at cols h*32.
__global__ __launch_bounds__(256) void attn_flash_kernel(
    const float* __restrict__ Q, const float* __restrict__ K,
    const float* __restrict__ V, float* __restrict__ O) {
  __shared__ float ldsK[32][36];      // 144B row stride: 16-bank column reads
  __shared__ float ldsV[32][36];
  __shared__ float ldsP[8][16][36];   // per-wave P staging (C->A layout bridge)

  const int bh = blockIdx.x;
  const int b = bh / N_HEADS, h = bh % N_HEADS;
  const int tid = threadIdx.x;
  const int wave = tid / 32, lane = tid % 32;
  const int l16 = lane % 16, half = lane / 16;
  const int qbase = blockIdx.y * 128 + wave * 16;

  const float* Qb = Q + ((size_t)b * SEQ) * HIDDEN + h * HEAD_D;
  const float* Kb = K + ((size_t)b * SEQ) * HIDDEN + h * HEAD_D;
  const float* Vb = V + ((size_t)b * SEQ) * HIDDEN + h * HEAD_D;

  // Preload Q tile as 8 A-fragments (16x4 each, covering d=0..31)
  v2f aq[8];
  {
    const int m = qbase + l16;
#pragma unroll
    for (int kk = 0; kk < 8; ++kk) {
      const int d0 = kk * 4 + half * 2;
      const float2 qv = *(const float2*)(Qb + (size_t)m * HIDDEN + d0);
      aq[kk].x = qv.x; aq[kk].y = qv.y;
    }
  }

  v8f o0 = {}, o1 = {};
  float rm[8], rs[8];
#pragma unroll
  for (int i = 0; i < 8; ++i) { rm[i] = -1e30f; rs[i] = 0.f; }

  const int krow = tid / 8, kc4 = (tid % 8) * 4;
  for (int t = 0; t < SEQ / 32; ++t) {
    // ---- cooperative K/V tile load: 32 keys x 32 d (async -> LDS) ----
    ASYNC_CP16(Kb + (size_t)(t * 32 + krow) * HIDDEN + kc4, &ldsK[krow][kc4]);
    ASYNC_CP16(Vb + (size_t)(t * 32 + krow) * HIDDEN + kc4, &ldsV[krow][kc4]);
    ASYNC_WAIT();
    __syncthreads();

    // ---- scores: two 16-key subtiles, S = Q * K^T ----
    v8f s0 = {}, s1 = {};
#pragma unroll
    for (int kk = 0; kk < 8; ++kk) {
      const int d0 = kk * 4 + half * 2;
      v2f b0, b1;
      b0.x = ldsK[l16][d0];        b0.y = ldsK[l16][d0 + 1];
      b1.x = ldsK[16 + l16][d0];   b1.y = ldsK[16 + l16][d0 + 1];
      s0 = WMMA_F32(aq[kk], b0, s0);
      s1 = WMMA_F32(aq[kk], b1, s1);
    }

    // ---- online softmax (rows live in (vgpr i, lane-half); reduce 16 lanes) --
    v8f p0, p1;
#pragma unroll
    for (int i = 0; i < 8; ++i) {
      float mx = fmaxf(s0[i], s1[i]);
#pragma unroll
      for (int off = 1; off < 16; off <<= 1) mx = fmaxf(mx, __shfl_xor(mx, off, 32));
      const float nm = fmaxf(rm[i], mx);
      const float e0 = __expf(s0[i] - nm);
      const float e1 = __expf(s1[i] - nm);
      float sum = e0 + e1;
#pragma unroll
      for (int off = 1; off < 16; off <<= 1) sum += __shfl_xor(sum, off, 32);
      const float corr = __expf(rm[i] - nm);
      rs[i] = rs[i] * corr + sum;
      rm[i] = nm;
      p0[i] = e0; p1[i] = e1;
      o0[i] *= corr; o1[i] *= corr;
    }

    // ---- restage P (C layout) into LDS so it can be read as A fragments ----
#pragma unroll
    for (int i = 0; i < 8; ++i) {
      ldsP[wave][i + 8 * half][l16]      = p0[i];
      ldsP[wave][i + 8 * half][16 + l16] = p1[i];
    }
    // per-wave region; LDS ops are in-order within a wave

    // ---- O += P @ Vtile ----
#pragma unroll
    for (int kk = 0; kk < 8; ++kk) {
      const int k0 = kk * 4 + half * 2;
      v2f ap, bv0, bv1;
      ap.x  = ldsP[wave][l16][k0];   ap.y  = ldsP[wave][l16][k0 + 1];
      bv0.x = ldsV[k0][l16];         bv0.y = ldsV[k0 + 1][l16];
      bv1.x = ldsV[k0][16 + l16];    bv1.y = ldsV[k0 + 1][16 + l16];
      o0 = WMMA_F32(ap, bv0, o0);
      o1 = WMMA_F32(ap, bv1, o1);
    }
    __syncthreads();   // before next iteration overwrites K/V tiles
  }

  // ---- finalize: O /= rowsum, scatter into conc[b,s,h*32+d] ----
#pragma unroll
  for (int i = 0; i < 8; ++i) {
    const int row = qbase + i + 8 * half;
    const float inv = 1.f / rs[i];
    O[((size_t)b * SEQ + row) * HIDDEN + h * HEAD_D + l16]      = o0[i] * inv;
    O[((size_t)b * SEQ + row) * HIDDEN + h * HEAD_D + 16 + l16] = o1[i] * inv;
  }
}

// ======================= host orchestration ===============================
static inline void run_gemm(const float* A, const float* B, float* C,
                            int M, int N, int K, int epi, const float* bias,
                            float scaleval, hipStream_t stream) {
  dim3 grid((N + 127) / 128, M / 64);
  gemm_wmma_kernel<<<grid, 256, 0, stream>>>(A, B, C, M, N, K, epi, bias, scaleval);
}

extern "C" void kernel_launch(void* const* d_in, const int* in_sizes, int n_in,
                              void* d_out, int out_size, void* d_ws, size_t ws_size,
                              hipStream_t stream) {
  const float* x_image     = (const float*)d_in[0];
  const float* input_proj  = (const float*)d_in[1];
  const float* output_proj = (const float*)d_in[2];
  const float* p_e_q       = (const float*)d_in[3];
  const float* p_e_k       = (const float*)d_in[4];
  const float* p_e_v       = (const float*)d_in[5];
  const float* p_e_c       = (const float*)d_in[6];
  const float* p_e_ff1     = (const float*)d_in[7];
  const float* p_e_ff2     = (const float*)d_in[8];
  const float* b_e_ff1     = (const float*)d_in[9];
  const float* b_e_ff2     = (const float*)d_in[10];
  const float* e_o_bias    = (const float*)d_in[11];
  const float* e_o_scale   = (const float*)d_in[12];
  const float* b_e_bias_1  = (const float*)d_in[13];
  const float* b_e_bias_2  = (const float*)d_in[14];
  const float* b_e_scale_1 = (const float*)d_in[15];
  const float* b_e_scale_2 = (const float*)d_in[16];
  const float* e_pos_emb   = (const float*)d_in[17];
  const float* b_focal     = (const float*)d_in[18];

  float* ws = (float*)d_ws;
  const size_t BUF = (size_t)MTOT * HIDDEN;   // 2,097,152 floats
  float* xe   = ws;               // x_encode (persists)
  float* li   = ws + 1 * BUF;     // layer_input
  float* lin  = ws + 2 * BUF;     // layer_in / final enc_outputs
  float* qb   = ws + 3 * BUF;     // q        -> multi_head
  float* kb   = ws + 4 * BUF;     // k        -> ff2
  float* vb   = ws + 5 * BUF;     // v        -> x_heads
  float* conc = ws + 6 * BUF;     // attention output
  float* ff1  = ws + 7 * BUF;     // ff1 [8192,1024]; also patches [8192,192]

  const float scale_q = 0.17677669529663689f;  // 1/sqrt(32)

  // 1) patchify + input projection
  {
    const int total = BATCH * SEQ * IN_CH;
    patchify_kernel<<<(total + 255) / 256, 256, 0, stream>>>(ff1, x_image);
    run_gemm(ff1, input_proj, xe, MTOT, HIDDEN, IN_CH, 0, nullptr, 0.f, stream);
  }

  const float* cur_in = xe;
  for (int m = 0; m < N_LAYERS; ++m) {
    const float* pos = e_pos_emb + (size_t)m * SEQ * HIDDEN;
    // layer_in = layer_input + pos (broadcast over batch)
    add_pos_kernel<<<(MTOT * HIDDEN / 4) / 256, 256, 0, stream>>>(lin, cur_in, pos);
    // Q (pre-scaled), K, V
    run_gemm(lin, p_e_q + (size_t)m * HIDDEN * HIDDEN, qb, MTOT, HIDDEN, HIDDEN, 1, nullptr, scale_q, stream);
    run_gemm(lin, p_e_k + (size_t)m * HIDDEN * HIDDEN, kb, MTOT, HIDDEN, HIDDEN, 0, nullptr, 0.f, stream);
    run_gemm(lin, p_e_v + (size_t)m * HIDDEN * HIDDEN, vb, MTOT, HIDDEN, HIDDEN, 0, nullptr, 0.f, stream);
    // attention -> conc
    attn_flash_kernel<<<dim3(BATCH * N_HEADS, SEQ / 128), 256, 0, stream>>>(qb, kb, vb, conc);
    // multi_head = conc @ Wc  (reuse q buffer)
    run_gemm(conc, p_e_c + (size_t)m * HIDDEN * HIDDEN, qb, MTOT, HIDDEN, HIDDEN, 0, nullptr, 0.f, stream);
    // x_heads = layer_in + LN(multi_head)  (reuse v buffer)
    ln_add_kernel<<<MTOT / 8, 256, 0, stream>>>(vb, lin, qb,
        b_e_bias_1 + (size_t)m * HIDDEN, b_e_scale_1 + (size_t)m * HIDDEN);
    // ff1 = relu(x_heads @ W1 + b1)
    run_gemm(vb, p_e_ff1 + (size_t)m * HIDDEN * FFWD, ff1, MTOT, FFWD, HIDDEN, 2,
             b_e_ff1 + (size_t)m * FFWD, 0.f, stream);
    // ff2 = ff1 @ W2 + b2  (reuse k buffer)
    run_gemm(ff1, p_e_ff2 + (size_t)m * FFWD * HIDDEN, kb, MTOT, HIDDEN, FFWD, 3,
             b_e_ff2 + (size_t)m * HIDDEN, 0.f, stream);
    // layer_input = x_heads + LN(ff2)
    ln_add_kernel<<<MTOT / 8, 256, 0, stream>>>(li, vb, kb,
        b_e_bias_2 + (size_t)m * HIDDEN, b_e_scale_2 + (size_t)m * HIDDEN);
    cur_in = li;
  }

  // enc_outputs = x_encode + LN(layer_input)  (reuse lin)
  ln_add_kernel<<<MTOT / 8, 256, 0, stream>>>(lin, xe, li, e_o_bias, e_o_scale);
  // output projection with sigmoid(first 4) / +b_focal(rest)
  run_gemm(lin, output_proj, (float*)d_out, MTOT, NCOUT, HIDDEN, 4, b_focal, 0.f, stream);
}